// LocallyConnected_24077586661416
// MI455X (gfx1250) — compile-verified
//
#include <hip/hip_runtime.h>
#include <hip/hip_bf16.h>
#include <math.h>

#define NAG    32
#define BATCH_ 8192
#define SDIM   48
#define ADIM   10
#define GDIM   128
#define HID    36

typedef float v2f __attribute__((ext_vector_type(2)));
typedef float v8f __attribute__((ext_vector_type(8)));

// D = A(16x4,f32) * B(4x16,f32) + C(16x16,f32)   -> v_wmma_f32_16x16x4_f32
__device__ __forceinline__ v8f wmma4(v2f a, v2f b, v8f c) {
  return __builtin_amdgcn_wmma_f32_16x16x4_f32(
      /*neg_a=*/false, a, /*neg_b=*/false, b,
      /*c_mod=*/(short)0, c, /*reuse_a=*/false, /*reuse_b=*/false);
}

// B fragment from a row-major weight Wt[N][K]: B[k][ncol] = Wt[ncol][k].
// Lane holds ncol = lane&15, k = klo + {0,1}.  Columns >= nlim are zero-padded
// BRANCHLESSLY: clamp the address to row 0, then cndmask-select zero.
__device__ __forceinline__ v2f load_bfrag(const float* __restrict__ Wt, int K,
                                          int ncol, int nlim, int klo) {
  const bool ok = (ncol < nlim);
  const int  nc = ok ? ncol : 0;            // valid address for every lane
  v2f b;
  b.x = Wt[nc * K + klo];
  b.y = Wt[nc * K + klo + 1];
  b.x = ok ? b.x : 0.f;                     // v_cndmask, no exec-mask branch
  b.y = ok ? b.y : 0.f;
  return b;
}

// Branchless guarded bias load (broadcast along rows of the D tile).
__device__ __forceinline__ float load_bias(const float* __restrict__ p,
                                           int idx, int lim) {
  const bool ok = (idx < lim);
  const float v = p[ok ? idx : 0];
  return ok ? v : 0.f;
}

// Wave-uniform scalar read of a float (forces s_cbranch instead of exec mask).
__device__ __forceinline__ float uniform_load(const float* __restrict__ p) {
  return __int_as_float(__builtin_amdgcn_readfirstlane(__float_as_int(*p)));
}

// Dynamic-LDS layout (floats):
//   sf     [32][16][36]  agent state features for this batch tile
//   hscr   [8][16][36]   per-wave scratch (h1, then comm)
//   gX     [16][36]      global-feature tile
//   bveff  [32][36]      topo-weighted bv bias
#define SF_OFF   0
#define HS_OFF   (32 * 16 * HID)
#define GX_OFF   (HS_OFF + 8 * 16 * HID)
#define BV_OFF   (GX_OFF + 16 * HID)
#define SMEM_FLOATS (BV_OFF + NAG * HID)   // 24768 floats = 99072 bytes

__global__ __launch_bounds__(256)
void fused_macomm_kernel(const float* __restrict__ states,
                         const float* __restrict__ gstate,
                         const float* __restrict__ W1, const float* __restrict__ b1,
                         const float* __restrict__ W2, const float* __restrict__ b2,
                         const float* __restrict__ Wv, const float* __restrict__ bv,
                         const float* __restrict__ Wg, const float* __restrict__ bg,
                         const float* __restrict__ Wact, const float* __restrict__ bact,
                         const float* __restrict__ topo,
                         float* __restrict__ out) {
  extern __shared__ float smem[];
  float* sfL   = smem + SF_OFF;
  float* gX    = smem + GX_OFF;
  float* bveff = smem + BV_OFF;

  const int tid  = threadIdx.x;
  const int w    = tid >> 5;        // wave id 0..7
  const int lane = tid & 31;
  const int half = lane >> 4;       // 0: k-pair {0,1}, 1: k-pair {2,3}
  const int l16  = lane & 15;       // A row / B,C,D column within tile
  const int b0   = blockIdx.x * 16; // batch tile base
  float* hscr = smem + HS_OFF + w * 16 * HID;

  // ---------------- phase 0a: bveff[i][g] = sum_j topo[i,j] * bv[i,j,g] -----
  for (int idx = tid; idx < NAG * HID; idx += 256) {
    const int i = idx / HID, g = idx % HID;
    float s = 0.f;
    for (int j = 0; j < NAG; ++j)
      s += topo[i * NAG + j] * bv[(size_t)(i * NAG + j) * HID + g];
    bveff[i * HID + g] = s;
  }

  // ---------------- phase 0b: gX = GS . Wg^T + bg  (waves 0..2) -------------
  if (w < 3) {
    const int col0 = w * 16;
    const float bias = load_bias(bg, col0 + l16, HID);
    v8f acc;
    #pragma unroll
    for (int v = 0; v < 8; ++v) acc[v] = bias;
    const float* Ab = gstate + (size_t)b0 * GDIM;
    for (int ks = 0; ks < GDIM / 4; ++ks) {
      const int klo = ks * 4 + half * 2;
      v2f a; a.x = Ab[l16 * GDIM + klo]; a.y = Ab[l16 * GDIM + klo + 1];
      v2f b = load_bfrag(Wg, GDIM, col0 + l16, HID, klo);
      acc = wmma4(a, b, acc);
    }
    #pragma unroll
    for (int v = 0; v < 8; ++v) {
      const int m = v + half * 8;
      if (col0 + l16 < HID) gX[m * HID + col0 + l16] = acc[v];
    }
  }

  // ---------------- phase 1: per-agent state features sf (4 agents / wave) --
  for (int rep = 0; rep < 4; ++rep) {
    const int n = w + rep * 8;

    // layer 1: h1 = tanh(states . W1^T + b1), K = 48
    const float* Ab  = states + ((size_t)n * BATCH_ + b0) * SDIM;
    const float* W1p = W1 + (size_t)n * HID * SDIM;
    v8f acc[3];
    #pragma unroll
    for (int t = 0; t < 3; ++t) {
      const float bias = load_bias(b1 + n * HID, t * 16 + l16, HID);
      #pragma unroll
      for (int v = 0; v < 8; ++v) acc[t][v] = bias;
    }
    for (int ks = 0; ks < SDIM / 4; ++ks) {
      const int klo = ks * 4 + half * 2;
      v2f a; a.x = Ab[l16 * SDIM + klo]; a.y = Ab[l16 * SDIM + klo + 1];
      #pragma unroll
      for (int t = 0; t < 3; ++t) {
        v2f b = load_bfrag(W1p, SDIM, t * 16 + l16, HID, klo);
        acc[t] = wmma4(a, b, acc[t]);
      }
    }
    #pragma unroll
    for (int t = 0; t < 3; ++t) {
      const int c = t * 16 + l16;
      #pragma unroll
      for (int v = 0; v < 8; ++v) {
        const int m = v + half * 8;
        if (c < HID) hscr[m * HID + c] = tanhf(acc[t][v]);
      }
    }

    // layer 2: sf = tanh(h1 . W2^T + b2), K = 36
    const float* W2p = W2 + (size_t)n * HID * HID;
    #pragma unroll
    for (int t = 0; t < 3; ++t) {
      const float bias = load_bias(b2 + n * HID, t * 16 + l16, HID);
      #pragma unroll
      for (int v = 0; v < 8; ++v) acc[t][v] = bias;
    }
    for (int ks = 0; ks < HID / 4; ++ks) {
      const int klo = ks * 4 + half * 2;
      v2f a; a.x = hscr[l16 * HID + klo]; a.y = hscr[l16 * HID + klo + 1];
      #pragma unroll
      for (int t = 0; t < 3; ++t) {
        v2f b = load_bfrag(W2p, HID, t * 16 + l16, HID, klo);
        acc[t] = wmma4(a, b, acc[t]);
      }
    }
    float* sfn = sfL + (size_t)n * 16 * HID;
    #pragma unroll
    for (int t = 0; t < 3; ++t) {
      const int c = t * 16 + l16;
      #pragma unroll
      for (int v = 0; v < 8; ++v) {
        const int m = v + half * 8;
        if (c < HID) sfn[m * HID + c] = tanhf(acc[t][v]);
      }
    }
  }

  __syncthreads();   // sf (all agents), gX, bveff now visible to all waves

  // ---------------- phase 2: comm + action heads (4 agents / wave) ----------
  for (int rep = 0; rep < 4; ++rep) {
    const int i = w + rep * 8;

    // comm[i] = sum_j topo[i,j] * (sf[j] . Wv[i,j]^T), skip zero topo entries
    v8f acc[3];
    #pragma unroll
    for (int t = 0; t < 3; ++t)
      #pragma unroll
      for (int v = 0; v < 8; ++v) acc[t][v] = 0.f;

    for (int j = 0; j < NAG; ++j) {
      const float t_ij = uniform_load(topo + i * NAG + j);  // SGPR -> s_cbranch
      if (t_ij != 0.f) {
        const float* sfj = sfL + (size_t)j * 16 * HID;
        const float* Wvp = Wv + (size_t)(i * NAG + j) * HID * HID;
        for (int ks = 0; ks < HID / 4; ++ks) {
          const int klo = ks * 4 + half * 2;
          v2f a;
          a.x = t_ij * sfj[l16 * HID + klo];
          a.y = t_ij * sfj[l16 * HID + klo + 1];
          #pragma unroll
          for (int t = 0; t < 3; ++t) {
            v2f b = load_bfrag(Wvp, HID, t * 16 + l16, HID, klo);
            acc[t] = wmma4(a, b, acc[t]);
          }
        }
      }
    }

    // stash raw comm tile in per-wave scratch (D layout -> row-major)
    #pragma unroll
    for (int t = 0; t < 3; ++t) {
      const int c = t * 16 + l16;
      #pragma unroll
      for (int v = 0; v < 8; ++v) {
        const int m = v + half * 8;
        if (c < HID) hscr[m * HID + c] = acc[t][v];
      }
    }

    // action head: acts = tanh((comm + bveff + g) . Wact^T + bact), N=10 pad 16
    v8f ac;
    const float bias = load_bias(bact + i * ADIM, l16, ADIM);
    #pragma unroll
    for (int v = 0; v < 8; ++v) ac[v] = bias;
    const float* Wap = Wact + (size_t)i * ADIM * HID;
    const float* bvi = bveff + i * HID;
    for (int ks = 0; ks < HID / 4; ++ks) {
      const int klo = ks * 4 + half * 2;
      v2f a;
      a.x = hscr[l16 * HID + klo]     + bvi[klo]     + gX[l16 * HID + klo];
      a.y = hscr[l16 * HID + klo + 1] + bvi[klo + 1] + gX[l16 * HID + klo + 1];
      v2f b = load_bfrag(Wap, HID, l16, ADIM, klo);
      ac = wmma4(a, b, ac);
    }
    #pragma unroll
    for (int v = 0; v < 8; ++v) {
      const int m = v + half * 8;
      if (l16 < ADIM)
        out[(size_t)(b0 + m) * (NAG * ADIM) + i * ADIM + l16] = tanhf(ac[v]);
    }
  }
}

extern "C" void kernel_launch(void* const* d_in, const int* in_sizes, int n_in,
                              void* d_out, int out_size, void* d_ws, size_t ws_size,
                              hipStream_t stream) {
  (void)in_sizes; (void)n_in; (void)out_size; (void)d_ws; (void)ws_size;
  const float* states = (const float*)d_in[0];
  const float* gstate = (const float*)d_in[1];
  const float* W1     = (const float*)d_in[2];
  const float* b1     = (const float*)d_in[3];
  const float* W2     = (const float*)d_in[4];
  const float* b2     = (const float*)d_in[5];
  const float* Wv     = (const float*)d_in[6];
  const float* bv     = (const float*)d_in[7];
  const float* Wg     = (const float*)d_in[8];
  const float* bg     = (const float*)d_in[9];
  const float* Wact   = (const float*)d_in[10];
  const float* bact   = (const float*)d_in[11];
  const float* topo   = (const float*)d_in[12];
  float* out = (float*)d_out;

  const dim3 grid(BATCH_ / 16);   // 512 workgroups, one 16-row batch tile each
  const dim3 block(256);          // 8 wave32 waves
  const size_t smem = SMEM_FLOATS * sizeof(float);  // 99072 B dynamic LDS

  fused_macomm_kernel<<<grid, block, smem, stream>>>(
      states, gstate, W1, b1, W2, b2, Wv, bv, Wg, bg, Wact, bact, topo, out);
}